// MinimumSpanningTree_11982958756210
// MI455X (gfx1250) — compile-verified
//
#include <hip/hip_runtime.h>
#include <math.h>

// ---- problem constants (match reference) ----
constexpr int kH     = 128;
constexpr int kW     = 256;
constexpr int kN     = kH * kW;            // 32768 vertices
constexpr int kErow  = (kH - 1) * kW;      // 32512 row edges
constexpr int kEcol  = kH * (kW - 1);      // 32640 col edges
constexpr int kE     = kErow + kEcol;      // 65152 edges (== 16 * 4072)
constexpr int kB     = 4;                  // batch
constexpr int kC     = 64;                 // channels
constexpr int kIters = 16;

typedef float v2f __attribute__((ext_vector_type(2)));
typedef float v8f __attribute__((ext_vector_type(8)));

// Edge e -> endpoints (p,q), identical ordering to reference _build_edges():
// row edges first (i-major), then col edges (i-major over W-1).
__device__ __forceinline__ void edge_uv(int e, int& p, int& q) {
  if (e < kErow) {            // e = i*W + j, p = i*W + j = e, q = p + W
    p = e;
    q = e + kW;
  } else {
    int t = e - kErow;        // t = i*(W-1) + j
    int i = t / (kW - 1);
    int j = t - i * (kW - 1);
    p = i * kW + j;
    q = p + 1;
  }
}

#if defined(__HIP_DEVICE_COMPILE__) && __has_builtin(__builtin_amdgcn_wmma_f32_16x16x4_f32)
#define HAVE_WMMA_F32X4 1
#else
#define HAVE_WMMA_F32X4 0
#endif

// ---------------------------------------------------------------------------
// Kernel 1: edge weights via f32 WMMA (diag of Gram matrix).
// One wave handles 16 edges. For K-chunk t (4 channels), lane L (m=L&15,
// hi=L>>4) holds A[m][2hi], A[m][2hi+1] -- per the documented 16x4 f32 A
// layout -- which is simultaneously the B(4x16) layout for B = A^T, so we
// pass the same register pair for A and B. D's diagonal = per-edge dot
// products, accumulated over 16 WMMAs = 64 channels. Exact f32 math.
// ---------------------------------------------------------------------------
__global__ __launch_bounds__(256)
void edge_weight_wmma(const float* __restrict__ g, float* __restrict__ w) {
  const int b    = blockIdx.y;
  const int lane = threadIdx.x & 31;
  const int wave = threadIdx.x >> 5;
  const int tile = blockIdx.x * 8 + wave;      // 0..4071
  const int m    = lane & 15;
  const int hi   = lane >> 4;
  const int e    = tile * 16 + m;
  int p, q;
  edge_uv(e, p, q);
  const float* gb = g + (size_t)b * kC * kN;

#if HAVE_WMMA_F32X4
  v8f acc = {};
#pragma unroll
  for (int t = 0; t < 16; ++t) {
    const float* g0 = gb + (size_t)(4 * t + 2 * hi) * kN;   // channel c0
    v2f a;
    a.x = g0[p] - g0[q];                 // channel c0
    a.y = g0[kN + p] - g0[kN + q];       // channel c0+1
    // (neg_a, A, neg_b, B, c_mod, C, reuse_a, reuse_b)
    acc = __builtin_amdgcn_wmma_f32_16x16x4_f32(false, a, false, a,
                                                (short)0, acc, false, false);
  }
  // C/D layout: VGPR r holds D[8*hi + r][m]; the diagonal D[e][e] for this
  // tile's edge m lives in lane (m,hi) with hi == (m>=8), at r = m&7.
  const bool have = (hi == 0) ? (m < 8) : (m >= 8);
  if (have) {
    const int r = m & 7;
    float dot = (r == 0) ? acc[0] : (r == 1) ? acc[1] : (r == 2) ? acc[2]
              : (r == 3) ? acc[3] : (r == 4) ? acc[4] : (r == 5) ? acc[5]
              : (r == 6) ? acc[6] : acc[7];
    w[(size_t)b * kE + e] = sqrtf(dot);
  }
#else
  // VALU fallback (lanes m and m+16 duplicate the same edge; benign).
  float s = 0.f;
  for (int c = 0; c < kC; ++c) {
    float d = gb[(size_t)c * kN + p] - gb[(size_t)c * kN + q];
    s += d * d;
  }
  w[(size_t)b * kE + e] = sqrtf(s);
#endif
}

// ---------------------------------------------------------------------------
// Kernel 2: full 16-round Boruvka + compaction, one workgroup per batch.
// All state for batch b is touched only by block b, so __syncthreads()
// between phases is a full happens-before. Segment min uses u64 atomicMin
// on key = (float_bits(w)<<32)|eid, exactly the reference's (w, eid) order.
// ---------------------------------------------------------------------------
__global__ __launch_bounds__(1024)
void boruvka_mst(const float* __restrict__ w_all,
                 unsigned long long* __restrict__ minkey_all,
                 int* __restrict__ label_all,
                 int* __restrict__ succA_all,
                 int* __restrict__ succB_all,
                 int* __restrict__ sel_all,
                 int* __restrict__ out_all) {
  const int b   = blockIdx.x;
  const int tid = threadIdx.x;
  const int T   = 1024;

  const float*        w      = w_all      + (size_t)b * kE;
  unsigned long long* minkey = minkey_all + (size_t)b * kN;
  int*                label  = label_all  + (size_t)b * kN;
  int*                sA     = succA_all  + (size_t)b * kN;
  int*                sB     = succB_all  + (size_t)b * kN;
  int*                sel    = sel_all    + (size_t)b * kE;
  int*                out    = out_all    + (size_t)b * (kN - 1) * 2;

  // init (fill output with edge 0 = (0, 256), matching flatnonzero fill)
  for (int n = tid; n < kN; n += T) label[n] = n;
  for (int e = tid; e < kE; e += T) sel[e] = 0;
  for (int r = tid; r < kN - 1; r += T) { out[2 * r] = 0; out[2 * r + 1] = kW; }
  __syncthreads();

  for (int it = 0; it < kIters; ++it) {
    for (int n = tid; n < kN; n += T) minkey[n] = ~0ULL;
    __syncthreads();

    // per-component min edge (weight, eid) via u64 atomicMin
    for (int e = tid; e < kE; e += T) {
      int p, q;
      edge_uv(e, p, q);
      int cu = label[p], cv = label[q];
      if (cu != cv) {
        unsigned long long key =
            ((unsigned long long)__float_as_uint(w[e]) << 32) | (unsigned)e;
        atomicMin(&minkey[cu], key);
        atomicMin(&minkey[cv], key);
      }
    }
    __syncthreads();

    // hook + mark selected
    for (int n = tid; n < kN; n += T) {
      unsigned long long key = minkey[n];
      int s = n;
      if (key != ~0ULL) {
        int e = (int)(unsigned)key;
        int p, q;
        edge_uv(e, p, q);
        int cu = label[p], cv = label[q];
        s = (cu == n) ? cv : cu;     // other endpoint's component
        sel[e] = 1;
      }
      sA[n] = s;
    }
    __syncthreads();

    // break 2-cycles: keep the smaller index as root
    for (int n = tid; n < kN; n += T) {
      int s = sA[n];
      sB[n] = (sA[s] == n && n < s) ? n : s;
    }
    __syncthreads();

    // 16 pointer-jump doublings (ping-pong), exactly like the reference
    int* src = sB;
    int* dst = sA;
    for (int k = 0; k < kIters; ++k) {
      for (int n = tid; n < kN; n += T) dst[n] = src[src[n]];
      __syncthreads();
      int* tmp = src; src = dst; dst = tmp;
    }

    for (int n = tid; n < kN; n += T) label[n] = src[label[n]];
    __syncthreads();
  }

  // compact selected edges in original edge order -> out[b][r] = (u,v)
  __shared__ int sscan[1024];
  int running = 0;
  for (int base = 0; base < kE; base += T) {
    int e = base + tid;
    int flag = (e < kE) ? sel[e] : 0;
    sscan[tid] = flag;
    __syncthreads();
    for (int off = 1; off < T; off <<= 1) {
      int v = (tid >= off) ? sscan[tid - off] : 0;
      __syncthreads();
      sscan[tid] += v;
      __syncthreads();
    }
    int incl  = sscan[tid];
    int total = sscan[T - 1];
    if (flag) {
      int r = running + incl - 1;
      if (r < kN - 1) {
        int p, q;
        edge_uv(e, p, q);
        out[2 * r] = p;
        out[2 * r + 1] = q;
      }
    }
    running += total;
    __syncthreads();
  }
}

// ---------------------------------------------------------------------------
extern "C" void kernel_launch(void* const* d_in, const int* in_sizes, int n_in,
                              void* d_out, int out_size, void* d_ws, size_t ws_size,
                              hipStream_t stream) {
  const float* g = (const float*)d_in[0];   // (4, 64, 128, 256) f32
  int* out = (int*)d_out;                   // (4, 32767, 2) int32

  // workspace carve-out (~4.6 MB), 8-byte aligned first
  char* ws = (char*)d_ws;
  size_t off = 0;
  unsigned long long* minkey = (unsigned long long*)(ws + off); off += (size_t)kB * kN * sizeof(unsigned long long);
  float* w   = (float*)(ws + off); off += (size_t)kB * kE * sizeof(float);
  int* label = (int*)(ws + off);   off += (size_t)kB * kN * sizeof(int);
  int* sA    = (int*)(ws + off);   off += (size_t)kB * kN * sizeof(int);
  int* sB    = (int*)(ws + off);   off += (size_t)kB * kN * sizeof(int);
  int* sel   = (int*)(ws + off);   off += (size_t)kB * kE * sizeof(int);
  (void)ws_size; (void)in_sizes; (void)n_in; (void)out_size;

  // 4072 tiles of 16 edges, 8 waves (tiles) per 256-thread block -> 509 blocks
  dim3 wgrid((kE / 16) / 8, kB, 1);
  edge_weight_wmma<<<wgrid, 256, 0, stream>>>(g, w);

  boruvka_mst<<<kB, 1024, 0, stream>>>(w, minkey, label, sA, sB, sel, out);
}